// ToRGBLayer_60060822667494
// MI455X (gfx1250) — compile-verified
//
#include <hip/hip_runtime.h>
#include <hip/hip_bf16.h>

typedef __attribute__((ext_vector_type(2))) float v2f;
typedef __attribute__((ext_vector_type(4))) float v4f;
typedef __attribute__((ext_vector_type(8))) float v8f;

#define NB    16
#define CIN   256
#define COUT  3
#define WDIM  512
#define HIN   128
#define HW    (HIN * HIN)      // 16384
#define HOUT  256
#define FW    13               // filter taps
#define LWP   260              // LDS row pitch (floats): 16B-aligned rows, banks 4m+c -> conflict-free

// ---------------------------------------------------------------------------
// Kernel 1: fold styles + conv weight into padded per-batch weight matrix
//   wmat[b][m][i] = (m < 3) ? conv_w[m][i] * styles[b][i] : 0
//   styles[b][i]  = (dot(w[b], aff_w[i]) / sqrt(512) + aff_b[i]) / sqrt(256)
// ---------------------------------------------------------------------------
__global__ __launch_bounds__(256) void fold_weights_kernel(
    const float* __restrict__ w,        // (16,512)
    const float* __restrict__ aff_w,    // (256,512)
    const float* __restrict__ aff_b,    // (256)
    const float* __restrict__ conv_w,   // (3,256,1,1) flat
    float* __restrict__ wmat)           // (16,16,256)
{
    const int b = blockIdx.x;
    const int i = threadIdx.x;
    const float* wb  = w + b * WDIM;
    const float* awi = aff_w + i * WDIM;
    float acc = 0.0f;
    #pragma unroll 8
    for (int j = 0; j < WDIM; ++j) acc = fmaf(wb[j], awi[j], acc);
    const float s = (acc * 0.04419417382415922f + aff_b[i]) * 0.0625f;
    float* dst = wmat + b * (16 * CIN) + i;
    #pragma unroll
    for (int m = 0; m < 16; ++m)
        dst[m * CIN] = (m < COUT) ? conv_w[m * CIN + i] * s : 0.0f;
}

// ---------------------------------------------------------------------------
// Kernel 2: modulated 1x1 conv via V_WMMA_F32_16X16X4_F32 + clamp
//   ymid[b][o][p] = clamp(sum_i x[b][i][p] * wmat[b][o][i] + bias[o], +-256)
// Per wave: D(16x16) += A(16x4 weights, LDS) * B(4x16 x-pixels, NT stream).
// Block = 256 threads = 8 waves = 8 pixel tiles; 128 blocks per batch.
// ---------------------------------------------------------------------------
__global__ __launch_bounds__(256) void modconv_wmma_kernel(
    const float* __restrict__ x,       // (16,256,128,128)
    const float* __restrict__ wmat,    // (16,16,256)
    const float* __restrict__ cbias,   // (3)
    float* __restrict__ ymid)          // (16,3,128,128)
{
    __shared__ __align__(16) float lW[16 * LWP];   // 16.25 KB
    const int tid  = threadIdx.x;
    const int wave = tid >> 5;
    const int lane = tid & 31;
    const int b        = blockIdx.x >> 7;          // 128 blocks per batch
    const int blk_in_b = blockIdx.x & 127;

    // Stage this batch's folded weights into LDS: b128 loads + b128 LDS stores
    const v4f* wsrc4 = (const v4f*)(wmat + b * (16 * CIN));
    for (int idx4 = tid; idx4 < (16 * CIN) / 4; idx4 += 256) {
        const int m  = idx4 >> 6;                  // 64 x float4 per 256-row
        const int c4 = (idx4 & 63) << 2;
        *(v4f*)&lW[m * LWP + c4] = wsrc4[idx4];    // m*1040 + c4*4: 16B aligned
    }
    __syncthreads();

    const int tile  = blk_in_b * 8 + wave;         // 0..1023
    const int pbase = tile * 16;
    const int n     = lane & 15;                   // pixel (B/N) & weight row (A/M)
    const int khalf = (lane >> 4) << 1;            // 0 or 2: K sub-column pair
    const float* xb = x + (size_t)b * CIN * HW + pbase + n;

    v8f acc = {};
    #pragma unroll 8
    for (int kk = 0; kk < CIN; kk += 4) {
        const int c0 = kk + khalf;
        // A fragment: lane holds A[M=n][K=c0..c0+1] -> single ds_load_b64
        v2f a = *(const v2f*)&lW[n * LWP + c0];
        // B fragment: x[b][c0][pixel], x[b][c0+1][pixel]; read-once 256MB
        // stream -> non-temporal so it doesn't thrash the 192MB L2
        v2f bv;
        bv.x = __builtin_nontemporal_load(xb + (size_t)c0 * HW);
        bv.y = __builtin_nontemporal_load(xb + (size_t)(c0 + 1) * HW);
        acc = __builtin_amdgcn_wmma_f32_16x16x4_f32(
            false, a, false, bv, (short)0, acc, false, false);
    }

    // D layout: VGPR r, lanes 0-15 -> row M=r (r=0..7). COUT=3 rows needed.
    if (lane < 16) {
        float* yb = ymid + (size_t)b * COUT * HW + pbase + n;
        #pragma unroll
        for (int o = 0; o < COUT; ++o) {
            float v = acc[o] + cbias[o];
            v = fminf(fmaxf(v, -256.0f), 256.0f);
            yb[(size_t)o * HW] = v;
        }
    }
}

// ---------------------------------------------------------------------------
// Kernel 3: horizontal zero-insert 2x upsample, 13-tap FIR (gain 2 per pass)
// ---------------------------------------------------------------------------
__global__ __launch_bounds__(256) void upsample_h_kernel(
    const float* __restrict__ ymid,    // (16,3,128,128)
    const float* __restrict__ filt,    // (13)
    float* __restrict__ yh)            // (16,3,128,256)
{
    const int idx = blockIdx.x * 256 + threadIdx.x;
    if (idx >= NB * COUT * HIN * HOUT) return;
    const int ox   = idx & 255;
    const int rest = idx >> 8;                 // b*3*128 + c*128 + h
    const float* row = ymid + (size_t)rest * HIN;
    float acc = 0.0f;
    #pragma unroll
    for (int t = 0; t < FW; ++t) {
        const int j = ox - 7 + t;              // dilated-input index
        if (j & 1) continue;                   // zero-inserted sample
        const int src = j >> 1;
        if (src < 0 || src >= HIN) continue;
        acc = fmaf(2.0f * filt[FW - 1 - t], row[src], acc);
    }
    yh[idx] = acc;
}

// ---------------------------------------------------------------------------
// Kernel 4: vertical pass -> final output (16,3,256,256)
// ---------------------------------------------------------------------------
__global__ __launch_bounds__(256) void upsample_v_kernel(
    const float* __restrict__ yh,      // (16,3,128,256)
    const float* __restrict__ filt,    // (13)
    float* __restrict__ out)           // (16,3,256,256)
{
    const int idx = blockIdx.x * 256 + threadIdx.x;
    if (idx >= NB * COUT * HOUT * HOUT) return;
    const int ox = idx & 255;
    const int oy = (idx >> 8) & 255;
    const int bc = idx >> 16;                  // 0..47
    const float* col = yh + (size_t)bc * (HIN * HOUT) + ox;
    float acc = 0.0f;
    #pragma unroll
    for (int t = 0; t < FW; ++t) {
        const int j = oy - 7 + t;
        if (j & 1) continue;
        const int src = j >> 1;
        if (src < 0 || src >= HIN) continue;
        acc = fmaf(2.0f * filt[FW - 1 - t], col[(size_t)src * HOUT], acc);
    }
    out[idx] = acc;
}

// ---------------------------------------------------------------------------
extern "C" void kernel_launch(void* const* d_in, const int* in_sizes, int n_in,
                              void* d_out, int out_size, void* d_ws, size_t ws_size,
                              hipStream_t stream) {
    const float* x      = (const float*)d_in[0];   // (16,256,128,128)
    const float* w      = (const float*)d_in[1];   // (16,512)
    const float* aff_w  = (const float*)d_in[2];   // (256,512)
    const float* aff_b  = (const float*)d_in[3];   // (256)
    const float* conv_w = (const float*)d_in[4];   // (3,256,1,1)
    const float* conv_b = (const float*)d_in[5];   // (3)
    const float* filt   = (const float*)d_in[6];   // (13)
    float* out = (float*)d_out;

    char* ws = (char*)d_ws;
    float* wmat = (float*)ws;                                   // 256 KB (16*16*256 floats)
    float* ymid = (float*)(ws + (64 << 10));                    // 3 MB
    float* yh   = (float*)(ws + (64 << 10) + (size_t)NB * COUT * HW * 4); // 6 MB

    // 1) fold weights
    fold_weights_kernel<<<NB, 256, 0, stream>>>(w, aff_w, aff_b, conv_w, wmat);

    // 2) WMMA modulated conv (+bias, clamp): 2048 blocks x 8 waves x 16 pixels
    modconv_wmma_kernel<<<NB * 128, 256, 0, stream>>>(x, wmat, conv_b, ymid);

    // 3) horizontal upsample: 16*3*128*256 elements
    upsample_h_kernel<<<(NB * COUT * HIN * HOUT) / 256, 256, 0, stream>>>(ymid, filt, yh);

    // 4) vertical upsample: 16*3*256*256 elements
    upsample_v_kernel<<<(NB * COUT * HOUT * HOUT) / 256, 256, 0, stream>>>(yh, filt, out);
}